// GATv2Layer4View_46411416600714
// MI455X (gfx1250) — compile-verified
//
#include <hip/hip_runtime.h>

#define BB 16
#define VV 4
#define NN 1024
#define FIN 128
#define HH 4
#define DD 32
#define HDIM 128
#define EE 131072
#define NTOT (BB*NN)          // 16384 global nodes
#define NEG_SLOPE 0.2f

typedef __attribute__((ext_vector_type(2))) float v2f;
typedef __attribute__((ext_vector_type(8))) float v8f;

// ---------------------------------------------------------------------------
// float atomic max via int/uint ordering trick (maps to global_atomic_max/min)
// ---------------------------------------------------------------------------
__device__ __forceinline__ void atomicMaxF(float* addr, float val) {
  if (val >= 0.0f) {
    atomicMax((int*)addr, __float_as_int(val));
  } else {
    atomicMin((unsigned int*)addr, (unsigned int)__float_as_uint(val));
  }
}

// ---------------------------------------------------------------------------
// B is W^T where W is [N,128] row-major: Beff[k][n] = W[n*128+k] (contiguous k)
// fp32 WMMA 16x16 tile, K=128 (V_WMMA_F32_16X16X4_F32).
// ---------------------------------------------------------------------------
__device__ __forceinline__ v8f wmma_k128_Bt(const float* __restrict__ aptr,
                                            const float* __restrict__ Wt,
                                            int bcol, int koff) {
  const float* bptr = Wt + (long)bcol * 128 + koff;
  v8f acc = {};
#pragma unroll
  for (int k = 0; k < 128; k += 4) {
    v2f a = *(const v2f*)(aptr + k);
    v2f b = *(const v2f*)(bptr + k);
    acc = __builtin_amdgcn_wmma_f32_16x16x4_f32(false, a, false, b, (short)0,
                                                acc, false, false);
  }
  return acc;
}

// ---------------------------------------------------------------------------
// Kernel 1: h1g / h2g = x @ W1 / W2, written in [V, NT, 128] layout.
// Per block: async-stage all of W (128x128 = 64KB) into LDS via
// global_load_async_to_lds_b128 (ASYNCcnt), then each of 8 waves computes a
// 16x128 strip (8 accumulators, A fragment reused 8x).  x is read exactly
// once from HBM.  grid (512, 1, 2), block 256.
// ---------------------------------------------------------------------------
__global__ void k_gemm_h(const float* __restrict__ x, const float* __restrict__ W1,
                         const float* __restrict__ W2, float* __restrict__ h1g,
                         float* __restrict__ h2g) {
  __shared__ float Wl[128 * 128];
  const float* W = (blockIdx.z == 0) ? W1 : W2;
  float* out = (blockIdx.z == 0) ? h1g : h2g;

  // ---- async copy W -> LDS (each thread moves 16 x b128 chunks) ----
  unsigned t = threadIdx.x;
  unsigned ldsbase = (unsigned)(unsigned long long)(&Wl[0]);
#pragma unroll
  for (int i = 0; i < 16; ++i) {
    unsigned off = (unsigned)(i * 256 + t) * 16u;   // byte offset
    unsigned ldsaddr = ldsbase + off;
    asm volatile("global_load_async_to_lds_b128 %0, %1, %2"
                 :: "v"(ldsaddr), "v"(off), "s"(W) : "memory");
  }
  asm volatile("s_wait_asynccnt 0x0" ::: "memory");
  __syncthreads();

  // ---- WMMA compute: one 16(M) x 128(N) strip per wave ----
  int wave = t >> 5;
  int lane = t & 31;
  int m0 = (blockIdx.x * 8 + wave) * 16;
  int koff = (lane >> 4) << 1;
  int r = m0 + (lane & 15);
  const float* aptr = x + (long)r * 128 + koff;
  int col = lane & 15;

  v8f acc[8];
  v8f zz = {};
#pragma unroll
  for (int j = 0; j < 8; ++j) acc[j] = zz;

  for (int k = 0; k < 128; k += 4) {
    v2f a = *(const v2f*)(aptr + k);
    const float* brow0 = &Wl[(k + koff) * 128 + col];
    const float* brow1 = &Wl[(k + koff + 1) * 128 + col];
#pragma unroll
    for (int j = 0; j < 8; ++j) {
      v2f b;
      b.x = brow0[j * 16];
      b.y = brow1[j * 16];
      acc[j] = __builtin_amdgcn_wmma_f32_16x16x4_f32(false, a, false, b,
                                                     (short)0, acc[j],
                                                     false, false);
    }
  }

  // ---- store with row remap r=(b*V+v)*N+n -> v*NT+b*N+n ----
  int half = (lane >> 4) << 3;
#pragma unroll
  for (int j = 0; j < 8; ++j) {
    int n0 = j * 16;
#pragma unroll
    for (int g = 0; g < 8; ++g) {
      int rr = m0 + half + g;
      int n = rr & (NN - 1);
      int bv = rr >> 10;
      int b = bv >> 2, v = bv & 3;
      out[((long)(v * NTOT + b * NN + n)) * 128 + n0 + col] = acc[j][g];
    }
  }
}

// ---------------------------------------------------------------------------
// init: segment-max to -inf, denom to 0   (V*NT*H = 262144 elements)
// ---------------------------------------------------------------------------
__global__ void k_init(float* __restrict__ mmax, float* __restrict__ denom) {
  int i = blockIdx.x * blockDim.x + threadIdx.x;
  if (i < VV * NTOT * HH) {
    mmax[i] = -__builtin_inff();
    denom[i] = 0.0f;
  }
}

// ---------------------------------------------------------------------------
// Pass A: per (view,edge,head) score + atomic segment-max over dst
// tid layout: h(2) | e(17) | v(2)  -> 2^21 threads
// ---------------------------------------------------------------------------
__global__ void k_edge_scores(const float* __restrict__ h1g, const float* __restrict__ h2g,
                              const int* __restrict__ src, const int* __restrict__ dst,
                              const float* __restrict__ ea, const float* __restrict__ U,
                              const float* __restrict__ att, float* __restrict__ scores,
                              float* __restrict__ mmax) {
  unsigned tid = blockIdx.x * blockDim.x + threadIdx.x;
  int h = tid & 3;
  int e = (tid >> 2) & (EE - 1);
  int v = tid >> 19;
  int s = src[e], d = dst[e];
  float a_e = ea[e];
  const float* p1 = h1g + ((long)(v * NTOT + s)) * 128 + h * 32;
  const float* p2 = h2g + ((long)(v * NTOT + d)) * 128 + h * 32;
  const float* pu = U + h * 32;
  const float* pa = att + h * 32;
  float acc = 0.0f;
#pragma unroll
  for (int k = 0; k < 32; k += 4) {
    float4 a1 = *(const float4*)(p1 + k);
    float4 a2 = *(const float4*)(p2 + k);
    float4 au = *(const float4*)(pu + k);
    float4 aa = *(const float4*)(pa + k);
    float s0 = a1.x + a2.x + a_e * au.x; s0 = (s0 > 0.0f) ? s0 : NEG_SLOPE * s0;
    float s1 = a1.y + a2.y + a_e * au.y; s1 = (s1 > 0.0f) ? s1 : NEG_SLOPE * s1;
    float s2 = a1.z + a2.z + a_e * au.z; s2 = (s2 > 0.0f) ? s2 : NEG_SLOPE * s2;
    float s3 = a1.w + a2.w + a_e * au.w; s3 = (s3 > 0.0f) ? s3 : NEG_SLOPE * s3;
    acc += aa.x * s0 + aa.y * s1 + aa.z * s2 + aa.w * s3;
  }
  scores[((long)v * EE + e) * 4 + h] = acc;
  atomicMaxF(&mmax[(v * NTOT + d) * 4 + h], acc);
}

// ---------------------------------------------------------------------------
// Pass B: ex = exp(score - max[dst]); denom[dst] += ex (in-place over scores)
// ---------------------------------------------------------------------------
__global__ void k_edge_softmax(const int* __restrict__ dst, const float* __restrict__ mmax,
                               float* __restrict__ scores, float* __restrict__ denom) {
  unsigned tid = blockIdx.x * blockDim.x + threadIdx.x;
  int h = tid & 3;
  int e = (tid >> 2) & (EE - 1);
  int v = tid >> 19;
  int d = dst[e];
  long si = ((long)v * EE + e) * 4 + h;
  float ex = __expf(scores[si] - mmax[(v * NTOT + d) * 4 + h]);
  scores[si] = ex;
  atomicAdd(&denom[(v * NTOT + d) * 4 + h], ex);
}

// ---------------------------------------------------------------------------
// Pass C: out[dst] += alpha * h1[src]   (float4 per thread, 4 atomic adds)
// tid layout: q(5) | e(17) | v(2)  -> 2^24 threads; q = float4 chunk (j = 4q)
// ---------------------------------------------------------------------------
__global__ void k_edge_aggregate(const float* __restrict__ h1g, const int* __restrict__ src,
                                 const int* __restrict__ dst, const float* __restrict__ scores,
                                 const float* __restrict__ denom, float* __restrict__ out) {
  unsigned tid = blockIdx.x * blockDim.x + threadIdx.x;
  int q = tid & 31;
  int e = (tid >> 5) & (EE - 1);
  int v = tid >> 22;
  int h = q >> 3;
  int s = src[e], d = dst[e];
  float alpha = scores[((long)v * EE + e) * 4 + h] /
                (denom[(v * NTOT + d) * 4 + h] + 1e-16f);
  const float4 hv = *(const float4*)(h1g + ((long)(v * NTOT + s)) * 128 + q * 4);
  float* po = out + ((long)(v * NTOT + d)) * 128 + q * 4;
  atomicAdd(po + 0, alpha * hv.x);
  atomicAdd(po + 1, alpha * hv.y);
  atomicAdd(po + 2, alpha * hv.z);
  atomicAdd(po + 3, alpha * hv.w);
}

// ---------------------------------------------------------------------------
// Kernel: qkv = tok @ in_proj_w^T + in_proj_b.  Logical A row r = nt*4+view
// maps to gatout row view*NT+nt.  Output [65536, 384].
// grid (512, 24), block 256
// ---------------------------------------------------------------------------
__global__ void k_gemm_qkv(const float* __restrict__ gatout, const float* __restrict__ ipw,
                           const float* __restrict__ ipb, float* __restrict__ qkv) {
  int wave = threadIdx.x >> 5;
  int lane = threadIdx.x & 31;
  int m0 = (blockIdx.x * 8 + wave) * 16;
  int n0 = blockIdx.y * 16;
  int koff = (lane >> 4) << 1;
  int r = m0 + (lane & 15);
  int nt = r >> 2, view = r & 3;
  const float* aptr = gatout + ((long)(view * NTOT + nt)) * 128 + koff;
  int bcol = n0 + (lane & 15);
  v8f acc = wmma_k128_Bt(aptr, ipw, bcol, koff);
  float badd = ipb[bcol];
  int half = (lane >> 4) << 3;
#pragma unroll
  for (int g = 0; g < 8; ++g) {
    qkv[(long)(m0 + half + g) * 384 + bcol] = acc[g] + badd;
  }
}

// ---------------------------------------------------------------------------
// Per (token, head) attention over the 4 views (tiny 4x4 softmax)
// ---------------------------------------------------------------------------
__global__ void k_attn(const float* __restrict__ qkv, float* __restrict__ aout) {
  int tid = blockIdx.x * blockDim.x + threadIdx.x;  // NT*H = 65536
  int h = tid & 3;
  int nt = tid >> 2;
  long base = (long)nt * 4;
  const float rs = 0.17677669529663687f;  // 1/sqrt(32)
  float logits[4][4];
#pragma unroll
  for (int i = 0; i < 4; ++i) {
    float qreg[32];
    const float* qp = qkv + (base + i) * 384 + h * 32;
#pragma unroll
    for (int d = 0; d < 32; ++d) qreg[d] = qp[d];
#pragma unroll
    for (int j = 0; j < 4; ++j) {
      const float* kp = qkv + (base + j) * 384 + 128 + h * 32;
      float s = 0.0f;
#pragma unroll
      for (int d = 0; d < 32; ++d) s += qreg[d] * kp[d];
      logits[i][j] = s * rs;
    }
  }
#pragma unroll
  for (int i = 0; i < 4; ++i) {
    float mx = fmaxf(fmaxf(logits[i][0], logits[i][1]),
                     fmaxf(logits[i][2], logits[i][3]));
    float w0 = __expf(logits[i][0] - mx);
    float w1 = __expf(logits[i][1] - mx);
    float w2 = __expf(logits[i][2] - mx);
    float w3 = __expf(logits[i][3] - mx);
    float inv = 1.0f / (w0 + w1 + w2 + w3);
    w0 *= inv; w1 *= inv; w2 *= inv; w3 *= inv;
    float o[32];
#pragma unroll
    for (int d = 0; d < 32; ++d) o[d] = 0.0f;
    float w[4] = {w0, w1, w2, w3};
#pragma unroll
    for (int j = 0; j < 4; ++j) {
      const float* vp = qkv + (base + j) * 384 + 256 + h * 32;
#pragma unroll
      for (int d = 0; d < 32; ++d) o[d] += w[j] * vp[d];
    }
    float* op = aout + (base + i) * 128 + h * 32;
#pragma unroll
    for (int d = 0; d < 32; ++d) op[d] = o[d];
  }
}

// ---------------------------------------------------------------------------
// Kernel: out = aout @ out_proj_w^T + out_proj_b + bias, scattered to
// [B,V,N,HD]: row r = nt*4+v -> b=nt>>10, n=nt&1023 -> ((b*4+v)*1024+n)*128+c
// grid (512, 8), block 256
// ---------------------------------------------------------------------------
__global__ void k_gemm_out(const float* __restrict__ aout, const float* __restrict__ opw,
                           const float* __restrict__ opb, const float* __restrict__ bias,
                           float* __restrict__ out) {
  int wave = threadIdx.x >> 5;
  int lane = threadIdx.x & 31;
  int m0 = (blockIdx.x * 8 + wave) * 16;
  int n0 = blockIdx.y * 16;
  int koff = (lane >> 4) << 1;
  int r = m0 + (lane & 15);
  const float* aptr = aout + (long)r * 128 + koff;
  int bcol = n0 + (lane & 15);
  v8f acc = wmma_k128_Bt(aptr, opw, bcol, koff);
  float badd = opb[bcol] + bias[bcol];
  int half = (lane >> 4) << 3;
#pragma unroll
  for (int g = 0; g < 8; ++g) {
    int rr = m0 + half + g;
    int nt = rr >> 2, view = rr & 3;
    int b = nt >> 10, n = nt & (NN - 1);
    out[((long)((b * 4 + view) * 1024 + n)) * 128 + bcol] = acc[g] + badd;
  }
}

// ---------------------------------------------------------------------------
extern "C" void kernel_launch(void* const* d_in, const int* in_sizes, int n_in,
                              void* d_out, int out_size, void* d_ws, size_t ws_size,
                              hipStream_t stream) {
  const float* x    = (const float*)d_in[0];
  const int*   ei   = (const int*)d_in[1];
  const float* ea   = (const float*)d_in[2];
  const float* W1   = (const float*)d_in[3];
  const float* W2   = (const float*)d_in[4];
  const float* U    = (const float*)d_in[5];
  const float* att  = (const float*)d_in[6];
  const float* ipw  = (const float*)d_in[7];
  const float* ipb  = (const float*)d_in[8];
  const float* opw  = (const float*)d_in[9];
  const float* opb  = (const float*)d_in[10];
  const float* bias = (const float*)d_in[11];
  float* out = (float*)d_out;

  const int* src = ei;
  const int* dst = ei + EE;

  const long SZ_H   = (long)VV * NTOT * 128;  // 8388608
  const long SZ_SC  = (long)VV * EE * HH;     // 2097152
  const long SZ_MD  = (long)VV * NTOT * HH;   //  262144

  float* ws     = (float*)d_ws;
  float* h1g    = ws;
  float* h2g    = h1g + SZ_H;
  float* gatout = h2g + SZ_H;                 // reused as attention output later
  float* scores = gatout + SZ_H;
  float* mmax   = scores + SZ_SC;
  float* denom  = mmax + SZ_MD;
  float* qkv    = denom + SZ_MD;
  float* aout   = gatout;                     // gatout dead after qkv GEMM

  // 1. projections h1g, h2g (WMMA fp32, async-LDS staged weights)
  k_gemm_h<<<dim3(512, 1, 2), 256, 0, stream>>>(x, W1, W2, h1g, h2g);

  // init accumulators
  hipMemsetAsync((void*)gatout, 0, SZ_H * sizeof(float), stream);
  k_init<<<(SZ_MD + 255) / 256, 256, 0, stream>>>(mmax, denom);

  // 2. GATv2 edge passes
  k_edge_scores<<<(VV * EE * HH) / 256, 256, 0, stream>>>(h1g, h2g, src, dst, ea,
                                                          U, att, scores, mmax);
  k_edge_softmax<<<(VV * EE * HH) / 256, 256, 0, stream>>>(dst, mmax, scores, denom);
  k_edge_aggregate<<<(VV * EE * 32) / 256, 256, 0, stream>>>(h1g, src, dst, scores,
                                                             denom, gatout);

  // 3. inter-view MHSA
  k_gemm_qkv<<<dim3(512, 24), 256, 0, stream>>>(gatout, ipw, ipb, qkv);
  k_attn<<<(NTOT * HH) / 256, 256, 0, stream>>>(qkv, aout);
  k_gemm_out<<<dim3(512, 8), 256, 0, stream>>>(aout, opw, opb, bias, out);
}